// QuantumLayer_12644383719616
// MI455X (gfx1250) — compile-verified
//
#include <hip/hip_runtime.h>
#include <stdint.h>

// MI455X / gfx1250 fused "quantum layer":
//   1) tiny setup kernel folds the 3 FFT-diagonal rotations into one real
//      64x64 circulant matrix M (the d_per-axis FFT cancels analytically).
//   2) main kernel: one workgroup (8 wave32) per (b,t); TDM-DMAs the 64x64
//      x-tile into LDS (tensor_load_to_lds, TENSORcnt), computes M @ x with
//      V_WMMA_F32_16X16X4_F32 (fp32 WMMA, no precision loss), then fuses
//      amplitude-noise + L2-normalize + |.| in one pass.
// Roofline: 128 MB HBM traffic @ 23.3 TB/s ~ 5.5us; 2.2 GFLOP is trivial for
// the WMMA pipe, so a single fused memory pass is optimal. Box-Muller noise
// is generated pairwise with fast-math transcendentals to keep the VALU
// epilogue under the memory roofline.

typedef __attribute__((ext_vector_type(2))) float        v2f;
typedef __attribute__((ext_vector_type(8))) float        v8f;
typedef __attribute__((ext_vector_type(4))) unsigned int v4u;
typedef __attribute__((ext_vector_type(4))) int          v4i;
typedef __attribute__((ext_vector_type(8))) int          v8i;

#define QD    64
#define DTOT  4096
#define NROT  16

// ---------------------------------------------------------------------------
// Kernel 1: build combined circulant rotation matrix M (64x64 f32) in d_ws.
//   theta_axis[q] = sum_r params[q][r];  a[j] = 0.5*theta[j], j<32
//   rx/ry diag:  d_k = cos(a[k>>1])                      (real)
//   rz    diag:  d_{2j} = e^{-i a_j},  d_{2j+1} = e^{+i a_j}
//   c_m = (1/64) sum_k d_k e^{+2pi i k m/64};  rotation = circulant(Re(c))
//   M = circ(cz) * circ(cy) * circ(cx)  ==> two circular convolutions.
// ---------------------------------------------------------------------------
__global__ __launch_bounds__(64) void build_rotation_matrix(
    const float* __restrict__ rx, const float* __restrict__ ry,
    const float* __restrict__ rz, float* __restrict__ Mout)
{
  __shared__ float ax[32], ay[32], az[32];
  __shared__ float cx[64], cy[64], cz[64], t1[64], tot[64];
  const int m = threadIdx.x;

  if (m < 32) {
    float sx = 0.f, sy = 0.f, sz = 0.f;
    for (int r = 0; r < NROT; ++r) {
      sx += rx[m * NROT + r];
      sy += ry[m * NROT + r];
      sz += rz[m * NROT + r];
    }
    ax[m] = 0.5f * sx; ay[m] = 0.5f * sy; az[m] = 0.5f * sz;
  }
  __syncthreads();

  const float w0 = (float)(6.283185307179586 / 64.0);
  float accx = 0.f, accy = 0.f, accz = 0.f;
  for (int j = 0; j < 32; ++j) {
    float phi0 = w0 * (float)((2 * j) * m);
    float phi1 = w0 * (float)((2 * j + 1) * m);
    float c0 = cosf(phi0), c1 = cosf(phi1);
    accx += cosf(ax[j]) * (c0 + c1);
    accy += cosf(ay[j]) * (c0 + c1);
    accz += cosf(phi0 - az[j]) + cosf(phi1 + az[j]);
  }
  cx[m] = accx * (1.f / 64.f);
  cy[m] = accy * (1.f / 64.f);
  cz[m] = accz * (1.f / 64.f);
  __syncthreads();

  float s = 0.f;
  for (int a = 0; a < 64; ++a) s += cy[a] * cx[(m - a) & 63];
  t1[m] = s;
  __syncthreads();
  s = 0.f;
  for (int a = 0; a < 64; ++a) s += cz[a] * t1[(m - a) & 63];
  tot[m] = s;
  __syncthreads();

  for (int k = 0; k < 64; ++k) Mout[m * 64 + k] = tot[(m - k) & 63];
}

// ---------------------------------------------------------------------------
// Deterministic counter-based gaussian pair (stand-in for jax threefry).
// Phase noise e^{i*ph} has |.|==1 and provably cannot affect the output.
// ---------------------------------------------------------------------------
__device__ __forceinline__ uint32_t hash_u32(uint32_t v) {
  v ^= 2747636419u; v *= 2654435769u;
  v ^= v >> 16;     v *= 2654435769u;
  v ^= v >> 16;     v *= 2654435769u;
  return v;
}
__device__ __forceinline__ void gauss_pair(uint32_t idx, float* n0, float* n1) {
  uint32_t h1 = hash_u32(idx * 2u + 1u);
  uint32_t h2 = hash_u32(idx * 2u + 2u);
  float u1 = ((float)h1 + 1.0f) * 2.3283064e-10f;        // (0,1]
  float u2 = (float)h2 * 2.3283064365386963e-10f;        // [0,1)
  float r  = sqrtf(-2.0f * __logf(u1));
  float sn, cs;
  __sincosf(6.2831853071795864f * u2, &sn, &cs);
  *n0 = r * cs;
  *n1 = r * sn;
}

// ---------------------------------------------------------------------------
// Kernel 2: per-(b,t) fused rotate + noise + normalize + measure.
//   out[q][d] = |(1+amp) * sum_k M[q][k] x[k][d]| / (||(1+amp)*s||_2 + 1e-8)
// 8 wave32 per block; wave w: row tile mi=w>>1, col tiles 2*(w&1)+{0,1}.
// x tile staged memory->LDS by the Tensor Data Mover (wave 0 issues the D#).
// ---------------------------------------------------------------------------
__global__ __launch_bounds__(256) void qlayer_main(
    const float* __restrict__ x, const float* __restrict__ M,
    float* __restrict__ out)
{
  __shared__ struct {
    float xs[QD * QD];   // 16 KB tile at LDS offset 0 (then reused for state)
    float red;
  } sm;
  float* xs = sm.xs;

  const int bt  = blockIdx.x;
  const int tid = threadIdx.x;
  const float* xp = x + (size_t)bt * DTOT;

  const int w    = tid >> 5;
  const int l    = tid & 31;
  const int mi   = w >> 1;          // 0..3 output row tile
  const int nb   = (w & 1) << 1;    // 0 or 2: col-tile base
  const int lr   = l & 15;
  const int kh   = l >> 4;          // half-select (0/1)
  const int row  = (mi << 4) + lr;  // output/M row for this lane's A data
  const int col0 = (nb << 4) + lr;  // B/D column for this lane

  // A operand: lane l holds M[row][4kc + 2*kh + {0,1}]  (contiguous pair)
  const float* Ma = M + row * 64 + (kh << 1);
  __builtin_prefetch(Ma, 0, 1);     // global_prefetch_b8: warm M while DMA runs

  if (tid == 0) sm.red = 0.f;

#if defined(__gfx1250__) && __has_builtin(__builtin_amdgcn_tensor_load_to_lds)
  // ---- Tensor Data Mover: DMA 4096 f32 (16 KB) memory -> LDS ----
  if (tid < 32) {                    // one issue per workgroup (EXEC ignored)
    const uint64_t ga  = (uint64_t)(uintptr_t)xp;
    const uint32_t lds = (uint32_t)(uintptr_t)xs;
    // D# group0: count=1 | lds_addr | global_addr[56:0] | type=2 (image)
    v4u g0 = { 1u,
               lds,
               (uint32_t)ga,
               (uint32_t)((ga >> 32) & 0x01FFFFFFu) | (2u << 30) };
    // D# group1: data_size=4B (code 2); 1-D tile: tile_dim0=4096,
    // tensor_dim0=4096, tensor_dim0_stride=4096; no multicast/pad/iterate.
    v8i g1 = { (int)(2u << 16),                      // [17:16] data_size
               (int)((4096u & 0xFFFFu) << 16),       // tensor_dim0 lo16
               (int)(4096u >> 16),                   // tensor_dim0 hi16
               (int)((4096u & 0xFFFFu) << 16),       // tile_dim0
               0,                                    // tile_dim1/2 unused
               (int)4096,                            // tensor_dim0_stride lo32
               0, 0 };
    v4i gz4 = { 0, 0, 0, 0 };
    v8i gz8 = { 0, 0, 0, 0, 0, 0, 0, 0 };
    // clang-23 / therock-10.0 6-arg form:
    //   (uint32x4 g0, int32x8 g1, int32x4, int32x4, int32x8, i32 cpol)
    __builtin_amdgcn_tensor_load_to_lds(g0, g1, gz4, gz4, gz8, 0);
    __builtin_amdgcn_s_wait_tensorcnt(0);
  }
  __syncthreads();
#else
  // Fallback: cooperative 128-bit staging.
  {
    const float4* xp4 = (const float4*)xp;
    float4* xs4 = (float4*)xs;
#pragma unroll
    for (int i = 0; i < 4; ++i) xs4[tid + 256 * i] = xp4[tid + 256 * i];
  }
  __syncthreads();
#endif

  v8f c0 = {}; v8f c1 = {};
#pragma unroll
  for (int kc = 0; kc < 16; ++kc) {
    v2f a;
    a.x = Ma[kc * 4 + 0];
    a.y = Ma[kc * 4 + 1];
    // B operand (4x16): VGPR v holds row (v + 2*kh) of the K-chunk.
    const int kr = (kc << 2) + (kh << 1);
    v2f b0, b1;
    b0.x = xs[(kr    ) * 64 + col0];
    b0.y = xs[(kr + 1) * 64 + col0];
    b1.x = xs[(kr    ) * 64 + col0 + 16];
    b1.y = xs[(kr + 1) * 64 + col0 + 16];
    // 8 args: (neg_a, A, neg_b, B, c_mod, C, reuse_a, reuse_b)
    c0 = __builtin_amdgcn_wmma_f32_16x16x4_f32(false, a, false, b0,
                                               (short)0, c0, false, false);
    c1 = __builtin_amdgcn_wmma_f32_16x16x4_f32(false, a, false, b1,
                                               (short)0, c1, false, false);
  }
  __syncthreads();  // all waves finished reading xs

  // Scatter D tiles back to LDS: VGPR j = rows (16mi + j) and (16mi + j + 8).
#pragma unroll
  for (int j = 0; j < 8; ++j) {
    const int r = (mi << 4) + j + (kh << 3);
    xs[r * 64 + col0]      = c0[j];
    xs[r * 64 + col0 + 16] = c1[j];
  }
  __syncthreads();

  // Amplitude noise (pairwise Box-Muller) + squared-sum reduction.
  const uint32_t base = (uint32_t)bt * DTOT;
  float v[16];
  float psum = 0.f;
#pragma unroll
  for (int p = 0; p < 8; ++p) {
    const int i0 = ((2 * p    ) << 8) + tid;   // coalesced: i*256 + tid
    const int i1 = ((2 * p + 1) << 8) + tid;
    float n0, n1;
    gauss_pair(base + (uint32_t)i0, &n0, &n1);
    const float v0 = (1.0f + 0.01f * n0) * xs[i0];
    const float v1 = (1.0f + 0.01f * n1) * xs[i1];
    v[2 * p]     = v0;
    v[2 * p + 1] = v1;
    psum += v0 * v0 + v1 * v1;
  }
  atomicAdd(&sm.red, psum);       // ds_add_f32
  __syncthreads();

  const float rn = 1.0f / (sqrtf(sm.red) + 1e-8f);
  float* op = out + (size_t)bt * DTOT;
#pragma unroll
  for (int i = 0; i < 16; ++i) {
    op[(i << 8) + tid] = fabsf(v[i]) * rn;
  }
}

// ---------------------------------------------------------------------------
extern "C" void kernel_launch(void* const* d_in, const int* in_sizes, int n_in,
                              void* d_out, int out_size, void* d_ws, size_t ws_size,
                              hipStream_t stream) {
  const float* x  = (const float*)d_in[0];   // (8, 512, 4096) f32
  const float* rx = (const float*)d_in[1];   // (64, 16) f32
  const float* ry = (const float*)d_in[2];
  const float* rz = (const float*)d_in[3];
  float* out = (float*)d_out;                // (8, 512, 4096) f32
  float* Mws = (float*)d_ws;                 // 64*64 f32 = 16 KB scratch

  build_rotation_matrix<<<1, 64, 0, stream>>>(rx, ry, rz, Mws);
  qlayer_main<<<8 * 512, 256, 0, stream>>>(x, Mws, out);
}